// Pre_Norm_Transformer_28527172780644
// MI455X (gfx1250) — compile-verified
//
#include <hip/hip_runtime.h>
#include <hip/hip_bf16.h>
#include <math.h>

// ---------------- problem constants ----------------
#define BB   64
#define NN   512
#define DIMF 128
#define EE   16
#define GG   32
#define TOPK 4
#define HID  512
#define EPSV 1e-8f
// 0.8 - 0.6*exp(-0.3*1)
#define LAMBDA_INIT 0.3555090675909693f

typedef __attribute__((ext_vector_type(2))) float v2f;
typedef __attribute__((ext_vector_type(8))) float v8f;

// fp32 WMMA: D(16x16) = A(16x4) * B(4x16) + C, full fp32 (matches reference precision)
__device__ __forceinline__ v8f wmma4(v2f a, v2f b, v8f c) {
  return __builtin_amdgcn_wmma_f32_16x16x4_f32(false, a, false, b, (short)0, c,
                                               false, false);
}

// CDNA5 async global->LDS copy (ASYNCcnt-tracked, bypasses VGPRs).
// lds = byte offset into LDS (low 32 bits of generic pointer to __shared__).
__device__ __forceinline__ void async_b128(unsigned lds, const float* g) {
  asm volatile("global_load_async_to_lds_b128 %0, %1, off"
               :: "v"(lds), "v"(g) : "memory");
}
__device__ __forceinline__ void wait_async0() {
  asm volatile("s_wait_asynccnt 0x0" ::: "memory");
}
__device__ __forceinline__ unsigned lds_off(const void* p) {
  return (unsigned)(uintptr_t)p;  // LDS aperture: addr[31:0] == LDS address
}

// ============ Kernel 1: rmsnorm + routing (gate, top-4, renorm) ============
__global__ __launch_bounds__(256) void k_norm_route(
    const float* __restrict__ x, const float* __restrict__ bias,
    const float* __restrict__ attn_scale, const float* __restrict__ Wr,
    const float* __restrict__ br, float* __restrict__ xn,
    float* __restrict__ route, float* __restrict__ out_idx) {
  __shared__ float xl[8][DIMF];
  __shared__ float gl[8][EE];
  const int tid = threadIdx.x, lane = tid & 31, wave = tid >> 5;
  const int tok = blockIdx.x * 8 + wave;             // B*N = 32768 tokens
  const float* xb = x + (size_t)tok * DIMF;

  float v[4], ss = 0.f;
#pragma unroll
  for (int s = 0; s < 4; ++s) { v[s] = xb[lane + 32 * s]; ss += v[s] * v[s]; }
#pragma unroll
  for (int off = 16; off > 0; off >>= 1) ss += __shfl_xor(ss, off, 32);
  const float inv = 1.0f / (sqrtf(ss * (1.0f / DIMF)) + EPSV);
#pragma unroll
  for (int s = 0; s < 4; ++s) {
    const int j = lane + 32 * s;
    const float xv = attn_scale[j] * v[s] * inv;
    xl[wave][j] = xv;
    xn[(size_t)tok * DIMF + j] = xv;
  }
  __syncthreads();
  if (lane < EE) {
    float d = 0.f;
#pragma unroll 16
    for (int j = 0; j < DIMF; ++j) d += xl[wave][j] * Wr[j * EE + lane];
    d += br[lane];
    gl[wave][lane] = 1.0f / (1.0f + __expf(-d));
  }
  __syncthreads();
  if (lane == 0) {
    float g[EE]; bool used[EE];
    const float b0 = bias[0];
#pragma unroll
    for (int e = 0; e < EE; ++e) { g[e] = gl[wave][e]; used[e] = false; }
    float sum = 0.f;
    for (int t = 0; t < TOPK; ++t) {      // lax.top_k: descending, lowest idx wins ties
      int best = 0; float bv = -3.4e38f;
      for (int e = 0; e < EE; ++e)
        if (!used[e] && (g[e] + b0) > bv) { bv = g[e] + b0; best = e; }
      used[best] = true; sum += g[best];
      out_idx[(size_t)tok * TOPK + t] = (float)best;
    }
    const float isum = 1.0f / sum;
    for (int e = 0; e < EE; ++e)
      route[(size_t)tok * EE + e] = used[e] ? g[e] * isum : 0.f;
  }
}

// ============ Kernel 2: softmax over rows of length 32 (E1 -> A1) ============
__global__ __launch_bounds__(256) void k_softmax32(const float* __restrict__ src,
                                                   float* __restrict__ dst) {
  const int lane = threadIdx.x & 31, wave = threadIdx.x >> 5;
  const size_t row = (size_t)blockIdx.x * 8 + wave;   // 2*E*N = 16384 rows
  const float v = src[row * GG + lane];
  float mx = v;
#pragma unroll
  for (int off = 16; off > 0; off >>= 1) mx = fmaxf(mx, __shfl_xor(mx, off, 32));
  const float e = __expf(v - mx);
  float s = e;
#pragma unroll
  for (int off = 16; off > 0; off >>= 1) s += __shfl_xor(s, off, 32);
  dst[row * GG + lane] = e / s;
}

// ============ Kernel 3: softmax over rows of length 512 (E2 -> A2) ============
__global__ __launch_bounds__(256) void k_softmax512(const float* __restrict__ src,
                                                    float* __restrict__ dst) {
  const int lane = threadIdx.x & 31, wave = threadIdx.x >> 5;
  const size_t row = (size_t)blockIdx.x * 8 + wave;   // 2*E*G = 1024 rows
  const float* p = src + row * NN;
  float v[16], mx = -3.4e38f;
#pragma unroll
  for (int s = 0; s < 16; ++s) { v[s] = p[lane + 32 * s]; mx = fmaxf(mx, v[s]); }
#pragma unroll
  for (int off = 16; off > 0; off >>= 1) mx = fmaxf(mx, __shfl_xor(mx, off, 32));
  float sum = 0.f;
#pragma unroll
  for (int s = 0; s < 16; ++s) { v[s] = __expf(v[s] - mx); sum += v[s]; }
#pragma unroll
  for (int off = 16; off > 0; off >>= 1) sum += __shfl_xor(sum, off, 32);
  const float is = 1.0f / sum;
#pragma unroll
  for (int s = 0; s < 16; ++s) dst[row * NN + lane + 32 * s] = v[s] * is;
}

// ============ Kernel 4: Y[b] = A2cat[1024x512] @ xn[b][512x128] ============
// Workgroup: 32 q-rows x 128 cols. Wave: 16x32 (dual accumulator, shared A frag).
// A/B tiles staged with global_load_async_to_lds_b128 (ASYNCcnt).
__global__ __launch_bounds__(256) void k_ygemm(const float* __restrict__ A2,
                                               const float* __restrict__ xn,
                                               float* __restrict__ Y) {
  __shared__ __align__(16) float At[32 * 64];
  __shared__ __align__(16) float Bt[64 * 128];
  const int b = blockIdx.y, q0 = blockIdx.x * 32;
  const int tid = threadIdx.x, lane = tid & 31, wave = tid >> 5;
  const int m = lane & 15, kq = (lane >> 4) << 1, rbase = (lane >> 4) << 3;
  const int wr = wave >> 2, wc = wave & 3;
  const float* xb = xn + (size_t)b * NN * DIMF;
  const unsigned at0 = lds_off(At), bt0 = lds_off(Bt);
  v8f acc0 = {}, acc1 = {};
  for (int k0 = 0; k0 < NN; k0 += 64) {
    __syncthreads();
#pragma unroll
    for (int u = 0; u < 2; ++u) {  // A tile: rows q0..q0+31, cols k0..k0+63
      const int idx = tid * 4 + 1024 * u, r = idx >> 6, c = idx & 63;
      async_b128(at0 + idx * 4, A2 + (size_t)(q0 + r) * NN + k0 + c);
    }
#pragma unroll
    for (int u = 0; u < 8; ++u) {  // B tile: xn rows k0..k0+63, all 128 cols
      const int idx = (tid + 256 * u) * 4, r = idx >> 7, c = idx & 127;
      async_b128(bt0 + idx * 4, xb + (size_t)(k0 + r) * DIMF + c);
    }
    wait_async0();
    __syncthreads();
    const float* Aw = At + wr * 16 * 64;
    const float* Bw = Bt + wc * 32;
#pragma unroll
    for (int kk = 0; kk < 64; kk += 4) {
      v2f a, b0, b1;
      a.x  = Aw[m * 64 + kk + kq];
      a.y  = Aw[m * 64 + kk + kq + 1];
      b0.x = Bw[(kk + kq) * 128 + m];
      b0.y = Bw[(kk + kq + 1) * 128 + m];
      b1.x = Bw[(kk + kq) * 128 + 16 + m];
      b1.y = Bw[(kk + kq + 1) * 128 + 16 + m];
      acc0 = wmma4(a, b0, acc0);
      acc1 = wmma4(a, b1, acc1);
    }
  }
  float* yb = Y + ((size_t)b * 1024 + q0 + wr * 16) * DIMF + wc * 32;
#pragma unroll
  for (int r = 0; r < 8; ++r) {
    yb[(size_t)(rbase + r) * DIMF + m]      = acc0[r];
    yb[(size_t)(rbase + r) * DIMF + 16 + m] = acc1[r];
  }
}

// ===== Kernel 5: attn[b] = W[b] @ Y[b] + x ; W built on the fly in LDS =====
// Workgroup: 32 tokens x 128 cols. Wave: 16x32 dual accumulator. Y tile async.
__global__ __launch_bounds__(256) void k_attn(
    const float* __restrict__ x, const float* __restrict__ route,
    const float* __restrict__ A1, const float* __restrict__ Y,
    const float* __restrict__ lq1, const float* __restrict__ lk1,
    const float* __restrict__ lq2, const float* __restrict__ lk2,
    float* __restrict__ xres) {
  __shared__ __align__(16) float Wt[32 * 64];
  __shared__ __align__(16) float Yt[64 * 128];
  __shared__ float coefs[2];
  const int b = blockIdx.y, i0 = blockIdx.x * 32;
  const int tid = threadIdx.x, lane = tid & 31, wave = tid >> 5;
  const int m = lane & 15, kq = (lane >> 4) << 1, rbase = (lane >> 4) << 3;
  const int wr = wave >> 2, wc = wave & 3;
  const unsigned yt0 = lds_off(Yt);
  if (tid == 0) {
    float d1 = 0.f, d2 = 0.f;
    for (int j = 0; j < DIMF; ++j) { d1 += lq1[j] * lk1[j]; d2 += lq2[j] * lk2[j]; }
    coefs[0] = 1.0f;
    coefs[1] = -(__expf(d1) - __expf(d2) + LAMBDA_INIT);   // lambda_full
  }
  __syncthreads();
  const float c0v = coefs[0], c1v = coefs[1];
  v8f acc0 = {}, acc1 = {};
  for (int q0c = 0; q0c < 1024; q0c += 64) {
    __syncthreads();
    const float* yb = Y + ((size_t)b * 1024 + q0c) * DIMF;
#pragma unroll
    for (int u = 0; u < 8; ++u) {  // Y tile: rows q0c..q0c+63 (contiguous copy)
      const int idx = (tid + 256 * u) * 4;
      async_b128(yt0 + idx * 4, yb + idx);
    }
#pragma unroll
    for (int u = 0; u < 8; ++u) {  // W tile [32 tokens x 64 q], computed in VALU
      const int idx = tid + 256 * u, r = idx >> 6, qq = idx & 63;
      const int q = q0c + qq;
      const int ec = q >> 9, k = (q >> 5) & 15, d = q & 31;
      const float cf = ec ? c1v : c0v;
      const float rt = route[((size_t)b * NN + i0 + r) * EE + k];
      const float a1 = A1[(((size_t)(ec * EE + k)) * NN + (i0 + r)) * GG + d];
      Wt[r * 64 + qq] = cf * rt * a1;
    }
    wait_async0();
    __syncthreads();
    const float* Aw = Wt + wr * 16 * 64;
    const float* Bw = Yt + wc * 32;
#pragma unroll
    for (int kk = 0; kk < 64; kk += 4) {
      v2f a, b0, b1;
      a.x  = Aw[m * 64 + kk + kq];
      a.y  = Aw[m * 64 + kk + kq + 1];
      b0.x = Bw[(kk + kq) * 128 + m];
      b0.y = Bw[(kk + kq + 1) * 128 + m];
      b1.x = Bw[(kk + kq) * 128 + 16 + m];
      b1.y = Bw[(kk + kq + 1) * 128 + 16 + m];
      acc0 = wmma4(a, b0, acc0);
      acc1 = wmma4(a, b1, acc1);
    }
  }
  const float* xb = x + ((size_t)b * NN + i0 + wr * 16) * DIMF + wc * 32;
  float* xo = xres + ((size_t)b * NN + i0 + wr * 16) * DIMF + wc * 32;
#pragma unroll
  for (int r = 0; r < 8; ++r) {
    const size_t o0 = (size_t)(rbase + r) * DIMF + m;
    const size_t o1 = o0 + 16;
    xo[o0] = xb[o0] + acc0[r];
    xo[o1] = xb[o1] + acc1[r];
  }
}

// ===== Kernel 6: fused FFN: rmsnorm -> SwiGLU (2 GEMMs) -> proj -> residual =====
__global__ __launch_bounds__(256) void k_ffn(
    const float* __restrict__ xres, const float* __restrict__ ffn_scale,
    const float* __restrict__ W1, const float* __restrict__ b1,
    const float* __restrict__ W2, const float* __restrict__ b2,
    const float* __restrict__ W3, const float* __restrict__ b3,
    float* __restrict__ out) {
  __shared__ __align__(16) float Xr[16 * DIMF];
  __shared__ __align__(16) float Xn[16 * DIMF];
  __shared__ __align__(16) float Hs[16 * HID];
  __shared__ float inv[16];
  const int t0 = blockIdx.x * 16;  // tile over B*N tokens
  const int tid = threadIdx.x, lane = tid & 31, wave = tid >> 5;
  const int m = lane & 15, kq = (lane >> 4) << 1, rbase = (lane >> 4) << 3;
  const float* xb = xres + (size_t)t0 * DIMF;
#pragma unroll
  for (int u = 0; u < 2; ++u) {
    const int idx = (tid + 256 * u) * 4;
    *(float4*)(Xr + idx) = *(const float4*)(xb + idx);
  }
  __syncthreads();
  if (tid < 16) {
    float s = 0.f;
    for (int j = 0; j < DIMF; ++j) { const float v = Xr[tid * DIMF + j]; s += v * v; }
    inv[tid] = 1.0f / (sqrtf(s * (1.0f / DIMF)) + EPSV);
  }
  __syncthreads();
#pragma unroll
  for (int u = 0; u < 8; ++u) {
    const int idx = tid + 256 * u, r = idx >> 7, c = idx & 127;
    Xn[idx] = ffn_scale[c] * Xr[idx] * inv[r];
  }
  __syncthreads();
  // H = silu(Xn@W1 + b1) * (Xn@W2 + b2): each wave does 4 col-tiles of 16
#pragma unroll
  for (int j = 0; j < 4; ++j) {
    const int n0 = (wave * 4 + j) * 16;
    v8f a1 = {}, a2 = {};
#pragma unroll
    for (int kk = 0; kk < DIMF; kk += 4) {
      v2f a, bA, bB;
      a.x = Xn[m * DIMF + kk + kq];
      a.y = Xn[m * DIMF + kk + kq + 1];
      bA.x = W1[(size_t)(kk + kq) * HID + n0 + m];
      bA.y = W1[(size_t)(kk + kq + 1) * HID + n0 + m];
      bB.x = W2[(size_t)(kk + kq) * HID + n0 + m];
      bB.y = W2[(size_t)(kk + kq + 1) * HID + n0 + m];
      a1 = wmma4(a, bA, a1);
      a2 = wmma4(a, bB, a2);
    }
    const float bb1 = b1[n0 + m], bb2 = b2[n0 + m];
#pragma unroll
    for (int r = 0; r < 8; ++r) {
      const float u = a1[r] + bb1, v = a2[r] + bb2;
      Hs[(rbase + r) * HID + n0 + m] = (u / (1.0f + __expf(-u))) * v;
    }
  }
  __syncthreads();
  // out = Xr + H@W3 + b3 : each wave one 16-col tile of 128
  const int c0 = wave * 16;
  v8f acc = {};
  for (int kk = 0; kk < HID; kk += 4) {
    v2f a, bb;
    a.x = Hs[m * HID + kk + kq];
    a.y = Hs[m * HID + kk + kq + 1];
    bb.x = W3[(size_t)(kk + kq) * DIMF + c0 + m];
    bb.y = W3[(size_t)(kk + kq + 1) * DIMF + c0 + m];
    acc = wmma4(a, bb, acc);
  }
  float* ob = out + (size_t)t0 * DIMF + c0;
  const float b3v = b3[c0 + m];
#pragma unroll
  for (int r = 0; r < 8; ++r)
    ob[(size_t)(rbase + r) * DIMF + m] = Xr[(rbase + r) * DIMF + c0 + m] + acc[r] + b3v;
}

// ---------------- host launcher ----------------
extern "C" void kernel_launch(void* const* d_in, const int* in_sizes, int n_in,
                              void* d_out, int out_size, void* d_ws, size_t ws_size,
                              hipStream_t stream) {
  const float* x          = (const float*)d_in[0];
  const float* bias       = (const float*)d_in[1];
  const float* attn_scale = (const float*)d_in[2];
  const float* ffn_scale  = (const float*)d_in[3];
  const float* Wr         = (const float*)d_in[4];
  const float* br         = (const float*)d_in[5];
  const float* E1         = (const float*)d_in[6];
  const float* E2         = (const float*)d_in[7];
  const float* lq1        = (const float*)d_in[8];
  const float* lk1        = (const float*)d_in[9];
  const float* lq2        = (const float*)d_in[10];
  const float* lk2        = (const float*)d_in[11];
  // d_in[12] = outer_lambda: dead (w*I + (1-w)*I == I)
  const float* W1         = (const float*)d_in[13];
  const float* b1         = (const float*)d_in[14];
  const float* W2         = (const float*)d_in[15];
  const float* b2         = (const float*)d_in[16];
  const float* W3         = (const float*)d_in[17];
  const float* b3         = (const float*)d_in[18];

  float* ws = (float*)d_ws;
  const size_t T = (size_t)BB * NN;                 // 32768 tokens
  float* xn    = ws;                                //  4,194,304
  float* route = xn    + T * DIMF;                  //    524,288
  float* A1    = route + T * EE;                    //    524,288
  float* A2    = A1    + 2 * EE * NN * GG;          //    524,288
  float* Y     = A2    + 2 * EE * GG * NN;          //  8,388,608
  float* xres  = Y     + (size_t)BB * 1024 * DIMF;  //  4,194,304  (~74 MB total)

  float* out     = (float*)d_out;
  float* out_idx = out + T * DIMF;                  // topk_idx as floats

  k_norm_route<<<dim3(T / 8), dim3(256), 0, stream>>>(x, bias, attn_scale, Wr, br,
                                                      xn, route, out_idx);
  k_softmax32 <<<dim3(2 * EE * NN / 8), dim3(256), 0, stream>>>(E1, A1);
  k_softmax512<<<dim3(2 * EE * GG / 8), dim3(256), 0, stream>>>(E2, A2);
  k_ygemm<<<dim3(1024 / 32, BB), dim3(256), 0, stream>>>(A2, xn, Y);
  k_attn <<<dim3(NN / 32, BB),  dim3(256), 0, stream>>>(x, route, A1, Y,
                                                        lq1, lk1, lq2, lk2, xres);
  k_ffn  <<<dim3(T / 16), dim3(256), 0, stream>>>(xres, ffn_scale, W1, b1, W2, b2,
                                                  W3, b3, out);
}